// ItemModeling_26027501813837
// MI455X (gfx1250) — compile-verified
//
#include <hip/hip_runtime.h>
#include <math.h>

#define Bn 4096
#define Ln 200
#define Dn 64
#define WAVES_A 4

typedef float v2f __attribute__((ext_vector_type(2)));
typedef float v8f __attribute__((ext_vector_type(8)));

__device__ __forceinline__ void lds_fence() {
  // CDNA5 split dep-counter: make all prior DS ops visible before proceeding,
  // and stop the compiler from reordering LDS traffic across this point.
  asm volatile("s_wait_dscnt 0" ::: "memory");
}

__device__ __forceinline__ v8f wmma4(v2f a, v2f b, v8f c) {
  // D = A(16x4 f32) * B(4x16 f32) + C(16x16 f32)
  return __builtin_amdgcn_wmma_f32_16x16x4_f32(false, a, false, b, (short)0, c,
                                               false, false);
}

__device__ __forceinline__ void zero4(v8f c[4]) {
#pragma unroll
  for (int i = 0; i < 4; ++i)
#pragma unroll
    for (int r = 0; r < 8; ++r) c[i][r] = 0.0f;
}

// C[16x64] += A(16 x kdim, LDS, row-major stride lda) * W(kdim x 64, global row-major)
// A fragment: lane = {M=lane&15}, VGPR e + lane-half -> K = ks + 2*(lane>>4) + e
// B fragment: mirrored (lane -> N, same K slot mapping), so the K permutation
// cancels between A and B over the full-K accumulation.
__device__ __forceinline__ void gemm_acc(const float* A, int lda,
                                         const float* __restrict__ W, int kdim,
                                         int lane, v8f c[4]) {
  const int half = lane >> 4;
  const int lq = lane & 15;
  for (int ks = 0; ks < kdim; ks += 4) {
    const int kb = ks + 2 * half;
    v2f a = *(const v2f*)(A + lq * lda + kb);
    const float* wr = W + (size_t)kb * 64 + lq;
#pragma unroll
    for (int nt = 0; nt < 4; ++nt) {
      v2f b;
      b.x = wr[nt * 16];
      b.y = wr[64 + nt * 16];
      c[nt] = wmma4(a, b, c[nt]);
    }
  }
}

// Epilogue: C/D layout -> out[M][col], VGPR r: M = r + 8*(lane>=16), N = lane&15
__device__ __forceinline__ void epi(v8f c[4], const float* __restrict__ bias,
                                    bool do_relu, float* out, int ldo, int lane) {
  const int half = lane >> 4;
  const int lq = lane & 15;
#pragma unroll
  for (int nt = 0; nt < 4; ++nt) {
    float bb = bias[nt * 16 + lq];
#pragma unroll
    for (int r = 0; r < 8; ++r) {
      float v = c[nt][r] + bb;
      if (do_relu) v = fmaxf(v, 0.0f);
      out[(r + 8 * half) * ldo + nt * 16 + lq] = v;
    }
  }
}

// ---------------- Phase A: per-row MLPs + attention logits ----------------
__global__ __launch_bounds__(WAVES_A * 32) void phaseA(
    const int* __restrict__ i_idx, const int* __restrict__ iul,
    const int* __restrict__ iur, const float* __restrict__ embed_i,
    const float* __restrict__ embed_u, const float* __restrict__ embed_r,
    const float* __restrict__ gu_w0, const float* __restrict__ gu_b0,
    const float* __restrict__ gu_w1, const float* __restrict__ gu_b1,
    const float* __restrict__ gu_w2, const float* __restrict__ gu_b2,
    const float* __restrict__ att_w1, const float* __restrict__ att_b1,
    const float* __restrict__ att_w2, const float* __restrict__ att_b2,
    float* __restrict__ fjt_g, float* __restrict__ a_g) {
  __shared__ float smem[WAVES_A * 3072];  // per wave: x[16][128] + h[16][64]
  const int lane = threadIdx.x & 31;
  const int wave = threadIdx.x >> 5;
  const int half = lane >> 4;
  const int lq = lane & 15;
  float* bufX = smem + wave * 3072;  // 16 x 128
  float* bufH = bufX + 2048;         // 16 x 64
  const int tile = blockIdx.x * WAVES_A + wave;
  const int base = tile * 16;  // flat (b*L + l) row base

  // 1) gather x = [pt | er] into LDS (lanes 0-15: user emb, 16-31: rating emb)
  for (int row = 0; row < 16; ++row) {
    const int fr = base + row;
    const float* src = (half == 0) ? (embed_u + (size_t)iul[fr] * 64)
                                   : (embed_r + (size_t)iur[fr] * 64);
    *(float4*)(bufX + row * 128 + half * 64 + 4 * lq) =
        *(const float4*)(src + 4 * lq);
  }
  lds_fence();

  v8f c[4];
  // 2) h1 = relu(x @ gu_w0 + b0)          (K=128)
  zero4(c);
  gemm_acc(bufX, 128, gu_w0, 128, lane, c);
  epi(c, gu_b0, true, bufH, 64, lane);
  lds_fence();

  // 3) h2 = relu(h1 @ gu_w1 + b1) -> bufX cols 0..63
  zero4(c);
  gemm_acc(bufH, 64, gu_w1, 64, lane, c);
  epi(c, gu_b1, true, bufX, 128, lane);

  // 4) gather qj (item emb, broadcast per batch row) into bufX cols 64..127
  for (int row = 0; row < 16; ++row) {
    const int fr = base + row;
    const float* src = embed_i + (size_t)i_idx[fr / Ln] * 64;
    *(float2*)(bufX + row * 128 + 64 + 2 * lane) =
        *(const float2*)(src + 2 * lane);
  }
  lds_fence();

  // 5) fjt = h2 @ gu_w2 + b2 -> bufH, and spill to global scratch
  zero4(c);
  gemm_acc(bufX, 128, gu_w2, 64, lane, c);
  epi(c, gu_b2, false, bufH, 64, lane);
  lds_fence();
  for (int i = lane; i < 256; i += 32) {
    const int row = i >> 4, cq = i & 15;
    *(float4*)(fjt_g + (size_t)(base + row) * 64 + 4 * cq) =
        *(const float4*)(bufH + row * 64 + 4 * cq);
  }

  // 6) h_att = relu([qj | fjt] @ att_w1 + att_b1) -> bufX cols 0..63
  zero4(c);
  gemm_acc(bufX + 64, 128, att_w1, 64, lane, c);       // qj part (rows 0..63)
  gemm_acc(bufH, 64, att_w1 + 64 * 64, 64, lane, c);   // fjt part (rows 64..127)
  epi(c, att_b1, true, bufX, 128, lane);
  lds_fence();

  // 7) a = h_att @ att_w2 + att_b2   (64 -> 1 dot, split across lane halves)
  float p = 0.0f;
  const float* hr = bufX + lq * 128 + 32 * half;
  const float* w2p = att_w2 + 32 * half;
#pragma unroll 8
  for (int j = 0; j < 32; ++j) p += hr[j] * w2p[j];
  bufH[lane] = p;
  lds_fence();
  if (lane < 16) a_g[base + lane] = bufH[lane] + bufH[lane + 16] + att_b2[0];
}

// -------- Phase B: batch-axis softmax stats (max, 1/sum-exp) per l --------
__global__ __launch_bounds__(256) void phaseB(const float* __restrict__ a,
                                              float* __restrict__ smax,
                                              float* __restrict__ sinv) {
  __shared__ float red[256];
  const int l = blockIdx.x;
  const int t = threadIdx.x;
  float m = -1e30f;
  for (int b = t; b < Bn; b += 256) m = fmaxf(m, a[(size_t)b * Ln + l]);
  red[t] = m;
  __syncthreads();
  for (int s = 128; s > 0; s >>= 1) {
    if (t < s) red[t] = fmaxf(red[t], red[t + s]);
    __syncthreads();
  }
  const float mx = red[0];
  __syncthreads();
  float acc = 0.0f;
  for (int b = t; b < Bn; b += 256) acc += __expf(a[(size_t)b * Ln + l] - mx);
  red[t] = acc;
  __syncthreads();
  for (int s = 128; s > 0; s >>= 1) {
    if (t < s) red[t] += red[t + s];
    __syncthreads();
  }
  if (t == 0) {
    smax[l] = mx;
    sinv[l] = 1.0f / red[0];
  }
}

// ---- Phase C: weighted L-reduction + final 64x64 linear + ReLU -> zj ----
__global__ __launch_bounds__(256) void phaseC(
    const float* __restrict__ a, const float* __restrict__ smax,
    const float* __restrict__ sinv, const float* __restrict__ fjt,
    const float* __restrict__ lin_w, const float* __restrict__ lin_b,
    float* __restrict__ out) {
  __shared__ float wts[16 * Ln];       // softmax weights for this 16-batch tile
  __shared__ float part[4 * 16 * 64];  // per-group partial sums
  __shared__ float sbuf[16 * 64];      // reduced S = sum_l mu*fjt
  const int t = threadIdx.x;
  const int bb = blockIdx.x * 16;

  for (int i = t; i < 16 * Ln; i += 256) {
    const int bl = i / Ln, l = i - bl * Ln;
    wts[i] = __expf(a[(size_t)(bb + bl) * Ln + l] - smax[l]) * sinv[l];
  }
  __syncthreads();

  const int col = t & 63;
  const int g = t >> 6;  // 4 groups of 64 threads split the L loop
  float acc[16];
#pragma unroll
  for (int i = 0; i < 16; ++i) acc[i] = 0.0f;
  for (int l = g; l < Ln; l += 4) {
    const float* fp = fjt + ((size_t)bb * Ln + l) * 64 + col;
#pragma unroll
    for (int bl = 0; bl < 16; ++bl)
      acc[bl] += wts[bl * Ln + l] * fp[(size_t)bl * Ln * 64];
  }
#pragma unroll
  for (int bl = 0; bl < 16; ++bl) part[g * 1024 + bl * 64 + col] = acc[bl];
  __syncthreads();
  for (int i = t; i < 1024; i += 256)
    sbuf[i] = part[i] + part[1024 + i] + part[2048 + i] + part[3072 + i];
  __syncthreads();

  if (t < 32) {  // wave 0: zj = relu(S @ lin_w + lin_b), 16x64x64 WMMA GEMM
    v8f cc[4];
    zero4(cc);
    gemm_acc(sbuf, 64, lin_w, 64, t, cc);
    const int half = t >> 4, lq = t & 15;
#pragma unroll
    for (int nt = 0; nt < 4; ++nt) {
      const float bbv = lin_b[nt * 16 + lq];
#pragma unroll
      for (int r = 0; r < 8; ++r) {
        const float v = fmaxf(cc[nt][r] + bbv, 0.0f);
        out[(size_t)(bb + r + 8 * half) * 64 + nt * 16 + lq] = v;
      }
    }
  }
}

extern "C" void kernel_launch(void* const* d_in, const int* in_sizes, int n_in,
                              void* d_out, int out_size, void* d_ws,
                              size_t ws_size, hipStream_t stream) {
  (void)in_sizes; (void)n_in; (void)out_size; (void)ws_size;
  const int* i_idx = (const int*)d_in[0];
  const int* iul = (const int*)d_in[1];
  const int* iur = (const int*)d_in[2];
  const float* embed_i = (const float*)d_in[3];
  const float* embed_u = (const float*)d_in[4];
  const float* embed_r = (const float*)d_in[5];
  const float* gu_w0 = (const float*)d_in[6];
  const float* gu_b0 = (const float*)d_in[7];
  const float* gu_w1 = (const float*)d_in[8];
  const float* gu_b1 = (const float*)d_in[9];
  const float* gu_w2 = (const float*)d_in[10];
  const float* gu_b2 = (const float*)d_in[11];
  const float* att_w1 = (const float*)d_in[12];
  const float* att_b1 = (const float*)d_in[13];
  const float* att_w2 = (const float*)d_in[14];
  const float* att_b2 = (const float*)d_in[15];
  const float* lin_w = (const float*)d_in[16];
  const float* lin_b = (const float*)d_in[17];

  float* fjt = (float*)d_ws;                       // B*L*64 f32 (210 MB)
  float* abuf = fjt + (size_t)Bn * Ln * 64;        // B*L logits
  float* smax = abuf + (size_t)Bn * Ln;            // 200
  float* sinv = smax + 256;                        // 200

  const int tiles = (Bn * Ln) / 16;                // 51200
  phaseA<<<tiles / WAVES_A, WAVES_A * 32, 0, stream>>>(
      i_idx, iul, iur, embed_i, embed_u, embed_r, gu_w0, gu_b0, gu_w1, gu_b1,
      gu_w2, gu_b2, att_w1, att_b1, att_w2, att_b2, fjt, abuf);
  phaseB<<<Ln, 256, 0, stream>>>(abuf, smax, sinv);
  phaseC<<<Bn / 16, 256, 0, stream>>>(abuf, smax, sinv, fjt, lin_w, lin_b,
                                      (float*)d_out);
}